// DRSL2_70901320123268
// MI455X (gfx1250) — compile-verified
//
#include <hip/hip_runtime.h>
#include <hip/hip_bf16.h>

// ---------------------------------------------------------------------------
// MI455X (gfx1250) implementation.
// Memory-bound: 4096*50257*4B = 823MB single read -> ~35us floor @ 23.3TB/s.
// No matmul structure -> WMMA N/A. CDNA5 paths used: wave32 reductions,
// async global->LDS DMA (ASYNCcnt + s_wait_asynccnt) double-buffered pipeline,
// ds_load_b128 consumption, v_exp_f32/v_log_f32 trans-pipe math.
// ---------------------------------------------------------------------------

#define A_COEF 1.0f
#define B_COEF 0.001f

constexpr int   BATCH  = 4096;
constexpr int   NCLASS = 50257;
constexpr int   TPB    = 256;        // 8 waves (wave32)
constexpr int   NWAVE  = TPB / 32;
constexpr int   NCH    = 50;         // ceil(max_float4s_per_row(12564)/256)
constexpr int   TPB2   = 1024;       // 32 waves for final reduction
constexpr float LOG2E  = 1.44269504088896340736f;
constexpr float LN2    = 0.69314718055994530942f;

typedef int v4i __attribute__((ext_vector_type(4)));
typedef __attribute__((address_space(1))) v4i as1_v4i;   // global
typedef __attribute__((address_space(3))) v4i as3_v4i;   // LDS

#if __has_builtin(__builtin_amdgcn_global_load_async_to_lds_b128)
  #define HAS_ASYNC 1
#else
  #define HAS_ASYNC 0
#endif

#if __has_builtin(__builtin_amdgcn_exp2f)
  #define EXP2F(x) __builtin_amdgcn_exp2f(x)
#else
  #define EXP2F(x) exp2f(x)
#endif
#if __has_builtin(__builtin_amdgcn_logf)
  #define LOG2F(x) __builtin_amdgcn_logf(x)
#else
  #define LOG2F(x) log2f(x)
#endif

#if HAS_ASYNC
  #if __has_builtin(__builtin_amdgcn_s_wait_asynccnt)
    #define WAITA(n) __builtin_amdgcn_s_wait_asynccnt(n)
  #else
    #define WAITA(n) asm volatile("s_wait_asynccnt %0" :: "i"(n) : "memory")
  #endif
#else
  #define WAITA(n) do {} while (0)
#endif

__device__ __forceinline__ void async_cp16(const float4* g, float4* l) {
#if HAS_ASYNC
  // global_load_async_to_lds_b128 : per-lane 16B DMA, tracked by ASYNCcnt
  __builtin_amdgcn_global_load_async_to_lds_b128(
      (as1_v4i*)(g), (as3_v4i*)(l), 0, 0);
#else
  *l = *g;  // synchronous fallback (still correct)
#endif
}

// ---------------------------------------------------------------------------
// Kernel 1: one block per row. Online logsumexp + full-row top-5 (single pass),
// streamed through a double-buffered LDS staging area via async DMA.
// ---------------------------------------------------------------------------
__global__ __launch_bounds__(TPB) void DRSL2_row_kernel(
    const float* __restrict__ pred, const int* __restrict__ labels,
    float* __restrict__ out_logp, float* __restrict__ out_xmean)
{
  __shared__ float4 buf[2][TPB];       // 8 KB double buffer
  __shared__ float  red[NWAVE][8];

  const int row = blockIdx.x;
  const int tid = threadIdx.x;
  const float* rp = pred + (size_t)row * NCLASS;

  const int   label    = labels[row];
  const float labelVal = rp[label];

  // Scalar prologue to reach 16B alignment (rows are only 4B aligned).
  const int pro = ((16 - (int)((size_t)rp & 15)) & 15) >> 2;   // 0..3 floats
  const float* ap  = rp + pro;
  const int    nf  = NCLASS - pro;
  const int    nf4 = nf >> 2;
  const int    rem = nf & 3;
  const float4* g4 = (const float4*)ap;

  // online-lse state (base-2) + full-row top-5 (descending)
  float m = -__builtin_inff(), mL = -__builtin_inff(), s = 0.0f;
  float t0 = -__builtin_inff(), t1 = -__builtin_inff(), t2 = -__builtin_inff();
  float t3 = -__builtin_inff(), t4 = -__builtin_inff();

  auto ins5 = [&](float v) {
    float a0 = fmaxf(v, t0),  b0 = fminf(v, t0);
    float a1 = fmaxf(b0, t1), b1 = fminf(b0, t1);
    float a2 = fmaxf(b1, t2), b2 = fminf(b1, t2);
    float a3 = fmaxf(b2, t3), b3 = fminf(b2, t3);
    float a4 = fmaxf(b3, t4);
    t0 = a0; t1 = a1; t2 = a2; t3 = a3; t4 = a4;
  };
  auto upd_max = [&](float cm) {
    if (cm > m) {                       // rare after warm-up
      float nL = cm * LOG2E;
      s *= EXP2F(mL - nL);              // exp2(-inf)=0 handles first update
      m = cm; mL = nL;
    }
  };
  auto proc4 = [&](float4 v) {
    float cm = fmaxf(fmaxf(v.x, v.y), fmaxf(v.z, v.w));
    upd_max(cm);
    s += EXP2F(fmaf(v.x, LOG2E, -mL));
    s += EXP2F(fmaf(v.y, LOG2E, -mL));
    s += EXP2F(fmaf(v.z, LOG2E, -mL));
    s += EXP2F(fmaf(v.w, LOG2E, -mL));
    if (cm > t4) { ins5(v.x); ins5(v.y); ins5(v.z); ins5(v.w); }
  };
  auto proc1 = [&](float v) {
    upd_max(v);
    s += EXP2F(fmaf(v, LOG2E, -mL));
    ins5(v);
  };

  // All lanes always issue (clamped) so per-wave ASYNCcnt is deterministic.
  auto issue = [&](int c) {
    int gi = c * TPB + tid;
    gi = (gi < nf4) ? gi : (nf4 - 1);
    async_cp16(g4 + gi, &buf[c & 1][tid]);
  };

  issue(0);
  for (int c = 0; c < NCH; ++c) {
    if (c + 1 < NCH) { issue(c + 1); WAITA(1); }   // chunk c landed
    else             { WAITA(0); }
    asm volatile("" ::: "memory");
    int gi = c * TPB + tid;
    if (gi < nf4) proc4(buf[c & 1][tid]);
  }

  // scalar prologue/epilogue elements (done after loop: mL is finite here)
  if (tid == 0) {
    for (int i = 0; i < pro; ++i) proc1(rp[i]);
    const float* ep = ap + (nf4 << 2);
    for (int i = 0; i < rem; ++i) proc1(ep[i]);
  }

  // wave32 butterfly reduction of (m,s) and top-5
  for (int off = 16; off > 0; off >>= 1) {
    float mo = __shfl_xor(m, off, 32);
    float so = __shfl_xor(s, off, 32);
    float u0 = __shfl_xor(t0, off, 32);
    float u1 = __shfl_xor(t1, off, 32);
    float u2 = __shfl_xor(t2, off, 32);
    float u3 = __shfl_xor(t3, off, 32);
    float u4 = __shfl_xor(t4, off, 32);
    float mn = fmaxf(m, mo);
    s = s * EXP2F((m - mn) * LOG2E) + so * EXP2F((mo - mn) * LOG2E);
    m = mn;
    ins5(u0); ins5(u1); ins5(u2); ins5(u3); ins5(u4);
  }

  const int wid = tid >> 5, lane = tid & 31;
  if (lane == 0) {
    red[wid][0] = m;  red[wid][1] = s;
    red[wid][2] = t0; red[wid][3] = t1; red[wid][4] = t2;
    red[wid][5] = t3; red[wid][6] = t4;
  }
  __syncthreads();

  if (tid == 0) {
    for (int w = 1; w < NWAVE; ++w) {
      float mo = red[w][0], so = red[w][1];
      float mn = fmaxf(m, mo);
      s = s * EXP2F((m - mn) * LOG2E) + so * EXP2F((mo - mn) * LOG2E);
      m = mn;
      ins5(red[w][2]); ins5(red[w][3]); ins5(red[w][4]);
      ins5(red[w][5]); ins5(red[w][6]);
    }
    const float lse = m + LN2 * LOG2F(s);           // natural-log lse of row
    out_logp[row] = labelVal - lse;                 // logp at label

    // masked top-4 statistic from full-row top-5 + labelVal:
    //   drop labelVal if it is in the top-5, else drop t4; masked max is t1
    //   when the label was the row argmax.
    const float sum5  = t0 + t1 + t2 + t3 + t4;
    const float dropA = (labelVal >= t4) ? labelVal : t4;
    const float mmax  = (labelVal == t0) ? t1 : t0;
    out_xmean[row] = (sum5 - dropA - mmax) * (1.0f / 3.0f);
  }
}

// ---------------------------------------------------------------------------
// Kernel 2: single block; CE mean + batch-dim log-softmax of x_mean.
// ---------------------------------------------------------------------------
__global__ __launch_bounds__(TPB2) void DRSL2_final_kernel(
    const float* __restrict__ logp, const float* __restrict__ xm,
    float* __restrict__ out)
{
  __shared__ float smA[32], smB[32], smC[32];
  __shared__ float bc[3];
  const int tid = threadIdx.x;
  const int wid = tid >> 5, lane = tid & 31;

  float xv[4];
  float lp_s = 0.0f, xm_s = 0.0f, xmax = -__builtin_inff();
  for (int k = 0; k < 4; ++k) {
    int i = tid + k * TPB2;
    lp_s += logp[i];
    float v = xm[i];
    xv[k] = v; xm_s += v; xmax = fmaxf(xmax, v);
  }
  for (int off = 16; off > 0; off >>= 1) {
    lp_s += __shfl_xor(lp_s, off, 32);
    xm_s += __shfl_xor(xm_s, off, 32);
    xmax  = fmaxf(xmax, __shfl_xor(xmax, off, 32));
  }
  if (lane == 0) { smA[wid] = lp_s; smB[wid] = xm_s; smC[wid] = xmax; }
  __syncthreads();
  if (wid == 0) {
    float a = smA[lane], b = smB[lane], c = smC[lane];
    for (int off = 16; off > 0; off >>= 1) {
      a += __shfl_xor(a, off, 32);
      b += __shfl_xor(b, off, 32);
      c  = fmaxf(c, __shfl_xor(c, off, 32));
    }
    if (lane == 0) { bc[0] = a; bc[1] = b; bc[2] = c; }
  }
  __syncthreads();

  const float M = bc[2];
  float es = 0.0f;
  for (int k = 0; k < 4; ++k) es += EXP2F((xv[k] - M) * LOG2E);
  for (int off = 16; off > 0; off >>= 1) es += __shfl_xor(es, off, 32);
  if (lane == 0) smA[wid] = es;
  __syncthreads();

  if (tid == 0) {
    float S = 0.0f;
    for (int w = 0; w < 32; ++w) S += smA[w];
    const float lseb = M + LN2 * LOG2F(S);
    const float ce   = -bc[0] / (float)BATCH;
    const float x    = (float)BATCH * lseb - bc[1];
    out[0] = A_COEF * ce + B_COEF * x;
  }
}

// ---------------------------------------------------------------------------
extern "C" void kernel_launch(void* const* d_in, const int* in_sizes, int n_in,
                              void* d_out, int out_size, void* d_ws, size_t ws_size,
                              hipStream_t stream) {
  (void)in_sizes; (void)n_in; (void)out_size; (void)ws_size;
  const float* pred   = (const float*)d_in[0];
  const int*   labels = (const int*)d_in[1];
  float* ws   = (float*)d_ws;             // needs 2*4096 floats = 32 KB
  float* logp = ws;
  float* xmn  = ws + BATCH;

  hipLaunchKernelGGL(DRSL2_row_kernel, dim3(BATCH), dim3(TPB), 0, stream,
                     pred, labels, logp, xmn);
  hipLaunchKernelGGL(DRSL2_final_kernel, dim3(1), dim3(TPB2), 0, stream,
                     logp, xmn, (float*)d_out);
}